// GAT_14224931684647
// MI455X (gfx1250) — compile-verified
//
#include <hip/hip_runtime.h>
#include <hip/hip_bf16.h>

#define HID 128
#define NEG_SLOPE 0.2f

typedef __attribute__((ext_vector_type(2))) float v2f;
typedef __attribute__((ext_vector_type(8))) float v8f;

// ---- order-preserving float <-> uint mapping for atomic segment-max ----
__device__ __forceinline__ unsigned f2ord(float f) {
  unsigned u = __float_as_uint(f);
  return (u & 0x80000000u) ? ~u : (u | 0x80000000u);
}
__device__ __forceinline__ float ord2f(unsigned u) {
  u = (u & 0x80000000u) ? (u ^ 0x80000000u) : ~u;
  return __uint_as_float(u);
}

// ============================================================
// h = X @ W  via V_WMMA_F32_16X16X4_F32 (exact fp32, matches ref)
// One wave -> one 16x16 tile of H; 8 waves/block share a 16x128
// strip of X staged in LDS (stride 132 floats -> conflict-free)
// via GLOBAL_LOAD_ASYNC_TO_LDS_B128 (ASYNCcnt-tracked DMA).
// ============================================================
__global__ __launch_bounds__(256) void gat_gemm_wmma(
    const float* __restrict__ X, const float* __restrict__ W,
    float* __restrict__ H, int N) {
  __shared__ __align__(16) float xs[16 * 132];
  const int tile = blockIdx.x;
  const int row0 = tile * 16;  // N is a multiple of 16 (50000 = 3125*16)

  // async DMA of the 16x128 X strip into padded LDS: 512 x 16B chunks.
  // LDS offset = low 32 bits of the generic pointer (ISA 10.2 aperture rule).
  {
    const char* gbase = (const char*)(X + (size_t)row0 * HID);
    const unsigned lbase = (unsigned)(size_t)(&xs[0]);
    for (int i = threadIdx.x; i < 16 * (HID / 4); i += 256) {
      int r = i >> 5, c4 = i & 31;
      unsigned loff = lbase + (unsigned)(r * 132 + c4 * 4) * 4u;
      unsigned long long gaddr =
          (unsigned long long)(gbase + ((size_t)r * HID + c4 * 4) * 4);
      asm volatile("global_load_async_to_lds_b128 %0, %1, off"
                   :: "v"(loff), "v"(gaddr)
                   : "memory");
    }
    asm volatile("s_wait_asynccnt 0" ::: "memory");
  }
  if (tile + 1 < (int)gridDim.x)  // speculative prefetch of next strip
    __builtin_prefetch(X + (size_t)(row0 + 16) * HID + threadIdx.x * 8, 0, 3);
  __syncthreads();

  const int wave = threadIdx.x >> 5;   // 0..7 -> output column tile
  const int lane = threadIdx.x & 31;
  const int half = lane >> 4;          // 0: K pair {0,1}; 1: K pair {2,3}
  const int mr   = lane & 15;
  const int col  = wave * 16 + mr;

  v8f acc = {};
  #pragma unroll 4
  for (int k0 = 0; k0 < HID; k0 += 4) {
    const int ka = k0 + 2 * half;
    // A frag: 16x4, lane (half,mr) holds X[row0+mr][ka], X[row0+mr][ka+1]
    v2f a = *reinterpret_cast<const v2f*>(xs + mr * 132 + ka);
    // B frag: 4x16, v0 holds rows {ka}, v1 holds rows {ka+1} per half-group
    v2f b;
    b[0] = W[(size_t)ka * HID + col];
    b[1] = W[(size_t)(ka + 1) * HID + col];
    acc = __builtin_amdgcn_wmma_f32_16x16x4_f32(
        false, a, false, b, (short)0, acc, false, false);
  }

  // D layout: vgpr i -> row (row0 + i + half*8), column col
  const int orow0 = row0 + half * 8;
  #pragma unroll
  for (int i = 0; i < 8; ++i)
    H[(size_t)(orow0 + i) * HID + col] = acc[i];
}

// ============================================================
// a_s[n] = h[n,:] . att_src ; a_d[n] = h[n,:] . att_dst
// one wave32 per node, float4 lanes + shuffle reduction
// ============================================================
__global__ __launch_bounds__(256) void gat_attn(
    const float* __restrict__ H, const float* __restrict__ att_s,
    const float* __restrict__ att_d, float* __restrict__ a_s,
    float* __restrict__ a_d, int N) {
  int t = blockIdx.x * blockDim.x + threadIdx.x;
  int node = t >> 5;
  int lane = t & 31;
  if (node >= N) return;
  float4 hv = reinterpret_cast<const float4*>(H)[(size_t)node * 32 + lane];
  float4 sv = reinterpret_cast<const float4*>(att_s)[lane];
  float4 dv = reinterpret_cast<const float4*>(att_d)[lane];
  float ps = hv.x * sv.x + hv.y * sv.y + hv.z * sv.z + hv.w * sv.w;
  float pd = hv.x * dv.x + hv.y * dv.y + hv.z * dv.z + hv.w * dv.w;
  #pragma unroll
  for (int off = 16; off > 0; off >>= 1) {
    ps += __shfl_down(ps, off, 32);
    pd += __shfl_down(pd, off, 32);
  }
  if (lane == 0) { a_s[node] = ps; a_d[node] = pd; }
}

// per-layer init: maxv := -inf key, denom := 0, xout[n][c] := bias[c]
__global__ __launch_bounds__(256) void gat_init_layer(
    unsigned* __restrict__ maxv, float* __restrict__ denom,
    float* __restrict__ xout, const float* __restrict__ bias, int N) {
  int i = blockIdx.x * blockDim.x + threadIdx.x;
  if (i < N) { maxv[i] = 0u; denom[i] = 0.0f; }
  if (i < N * HID) xout[i] = bias[i & (HID - 1)];
}

// edge pass 1: logits (leaky-relu) + atomic segment-max over dst
__global__ __launch_bounds__(256) void gat_edge_max(
    const int* __restrict__ ei, const float* __restrict__ a_s,
    const float* __restrict__ a_d, float* __restrict__ logits,
    unsigned* __restrict__ maxv, int E, int Etot) {
  int e = blockIdx.x * blockDim.x + threadIdx.x;
  if (e >= Etot) return;
  int s, d;
  if (e < E) { s = ei[e]; d = ei[E + e]; } else { s = d = e - E; }  // self-loops
  float z = a_s[s] + a_d[d];
  z = (z > 0.0f) ? z : NEG_SLOPE * z;
  logits[e] = z;
  atomicMax(maxv + d, f2ord(z));
}

// edge pass 2: e = exp(logit - max[dst]); denom[dst] += e (in-place in logits)
__global__ __launch_bounds__(256) void gat_edge_exp(
    const int* __restrict__ ei, float* __restrict__ logits,
    const unsigned* __restrict__ maxv, float* __restrict__ denom,
    int E, int Etot) {
  int e = blockIdx.x * blockDim.x + threadIdx.x;
  if (e >= Etot) return;
  int d = (e < E) ? ei[E + e] : (e - E);
  float ev = __expf(logits[e] - ord2f(maxv[d]));
  logits[e] = ev;
  atomicAdd(denom + d, ev);
}

// edge pass 3: xout[dst] += alpha * h[src]; one wave32 per edge, float4/lane
__global__ __launch_bounds__(256) void gat_aggregate(
    const int* __restrict__ ei, const float* __restrict__ H,
    const float* __restrict__ ealpha, const float* __restrict__ denom,
    float* __restrict__ xout, int E, int Etot) {
  int t = blockIdx.x * blockDim.x + threadIdx.x;
  int e = t >> 5;
  int lane = t & 31;
  if (e >= Etot) return;
  int s, d;
  if (e < E) { s = ei[e]; d = ei[E + e]; } else { s = d = e - E; }
  float alpha = ealpha[e] / (denom[d] + 1e-16f);
  float4 hv = reinterpret_cast<const float4*>(H)[(size_t)s * 32 + lane];
  float* o = xout + (size_t)d * HID + lane * 4;
  atomicAdd(o + 0, hv.x * alpha);
  atomicAdd(o + 1, hv.y * alpha);
  atomicAdd(o + 2, hv.z * alpha);
  atomicAdd(o + 3, hv.w * alpha);
}

__global__ __launch_bounds__(256) void gat_zero(float* __restrict__ p, int n) {
  int i = blockIdx.x * blockDim.x + threadIdx.x;
  if (i < n) p[i] = 0.0f;
}

// global_add_pool: pooled[batch[n]] += x[n]
__global__ __launch_bounds__(256) void gat_pool(
    const float* __restrict__ X, const int* __restrict__ batch,
    float* __restrict__ pooled, int N) {
  int t = blockIdx.x * blockDim.x + threadIdx.x;
  int n = t >> 5, lane = t & 31;
  if (n >= N) return;
  float4 xv = reinterpret_cast<const float4*>(X)[(size_t)n * 32 + lane];
  int g = batch[n];
  float* p = pooled + (size_t)g * HID + lane * 4;
  atomicAdd(p + 0, xv.x);
  atomicAdd(p + 1, xv.y);
  atomicAdd(p + 2, xv.z);
  atomicAdd(p + 3, xv.w);
}

// BN (inference) + FC head: 64 graphs x 64 outputs
__global__ __launch_bounds__(64) void gat_head(
    const float* __restrict__ pooled, const float* __restrict__ gamma,
    const float* __restrict__ beta, const float* __restrict__ mean,
    const float* __restrict__ var, const float* __restrict__ fcw,
    const float* __restrict__ fcb, float* __restrict__ out) {
  __shared__ float hbn[HID];
  int g = blockIdx.x;
  int j = threadIdx.x;  // 0..63
  for (int c = j; c < HID; c += 64) {
    float p = pooled[(size_t)g * HID + c];
    hbn[c] = (p - mean[c]) * rsqrtf(var[c] + 1e-5f) * gamma[c] + beta[c];
  }
  __syncthreads();
  float acc = fcb[j];
  #pragma unroll 4
  for (int c = 0; c < HID; ++c) acc += hbn[c] * fcw[c * 64 + j];
  out[g * 64 + j] = acc;
}

extern "C" void kernel_launch(void* const* d_in, const int* in_sizes, int n_in,
                              void* d_out, int out_size, void* d_ws, size_t ws_size,
                              hipStream_t stream) {
  const float* x        = (const float*)d_in[0];
  const int*   ei       = (const int*)d_in[1];
  const int*   batch    = (const int*)d_in[2];
  const float* Ws       = (const float*)d_in[3];
  const float* att_src  = (const float*)d_in[4];
  const float* att_dst  = (const float*)d_in[5];
  const float* biases   = (const float*)d_in[6];
  const float* bn_gamma = (const float*)d_in[7];
  const float* bn_beta  = (const float*)d_in[8];
  const float* bn_mean  = (const float*)d_in[9];
  const float* bn_var   = (const float*)d_in[10];
  const float* fc_w     = (const float*)d_in[11];
  const float* fc_b     = (const float*)d_in[12];

  const int N    = in_sizes[0] / HID;   // 50000
  const int E    = in_sizes[1] / 2;     // 800000
  const int Etot = E + N;               // edges + self-loops
  const int NG   = 64;

  // workspace carve-out (~81 MB total, fp32)
  float* ws = (float*)d_ws;
  size_t off = 0;
  auto carve = [&](size_t n) {
    float* p = ws + off;
    off += (n + 63) & ~(size_t)63;
    return p;
  };
  float*    hbuf   = carve((size_t)N * HID);
  float*    bufA   = carve((size_t)N * HID);
  float*    bufB   = carve((size_t)N * HID);
  float*    a_s    = carve(N);
  float*    a_d    = carve(N);
  unsigned* maxv   = (unsigned*)carve(N);
  float*    denom  = carve(N);
  float*    ebuf   = carve(Etot);
  float*    pooled = carve((size_t)NG * HID);
  (void)ws_size; (void)n_in; (void)out_size;

  const float* xin = x;
  float* xout = bufA;
  for (int l = 0; l < 3; ++l) {
    const float* Wl = Ws + (size_t)l * HID * HID;
    gat_gemm_wmma<<<N / 16, 256, 0, stream>>>(xin, Wl, hbuf, N);
    gat_attn<<<(N * 32 + 255) / 256, 256, 0, stream>>>(
        hbuf, att_src + l * HID, att_dst + l * HID, a_s, a_d, N);
    gat_init_layer<<<(N * HID + 255) / 256, 256, 0, stream>>>(
        maxv, denom, xout, biases + l * HID, N);
    gat_edge_max<<<(Etot + 255) / 256, 256, 0, stream>>>(
        ei, a_s, a_d, ebuf, maxv, E, Etot);
    gat_edge_exp<<<(Etot + 255) / 256, 256, 0, stream>>>(
        ei, ebuf, maxv, denom, E, Etot);
    gat_aggregate<<<(Etot * 32 + 255) / 256, 256, 0, stream>>>(
        ei, hbuf, ebuf, denom, xout, E, Etot);
    xin = xout;
    xout = (l == 0) ? bufB : bufA;
  }

  gat_zero<<<(NG * HID + 255) / 256, 256, 0, stream>>>(pooled, NG * HID);
  gat_pool<<<(N * 32 + 255) / 256, 256, 0, stream>>>(xin, batch, pooled, N);
  gat_head<<<NG, 64, 0, stream>>>(pooled, bn_gamma, bn_beta, bn_mean, bn_var,
                                  fc_w, fc_b, (float*)d_out);
}